// SGBlock_1949915152400
// MI455X (gfx1250) — compile-verified
//
#include <hip/hip_runtime.h>
#include <math.h>

typedef float v2f __attribute__((ext_vector_type(2)));
typedef float v8f __attribute__((ext_vector_type(8)));

__device__ __forceinline__ v8f vz8() {
  v8f z = {0.f,0.f,0.f,0.f,0.f,0.f,0.f,0.f};
  return z;
}

#define WMMA(a,b,c) __builtin_amdgcn_wmma_f32_16x16x4_f32(false,(a),false,(b),(short)0,(c),false,false)

// ---- CDNA5 async global->LDS staging (ASYNCcnt), with portable fallback ----
#if __has_builtin(__builtin_amdgcn_global_load_async_to_lds_b128) && __has_builtin(__builtin_amdgcn_s_wait_asynccnt)
#define USE_ASYNC 1
#else
#define USE_ASYNC 0
#endif

#if USE_ASYNC
typedef int v4i_t __attribute__((vector_size(16)));
typedef __attribute__((address_space(1))) v4i_t* gptr4_t;
typedef __attribute__((address_space(3))) v4i_t* lptr4_t;
#endif

__device__ __forceinline__ void cp16(float* lds, const float* g) {
#if USE_ASYNC
  __builtin_amdgcn_global_load_async_to_lds_b128(
      (gptr4_t)(v4i_t*)g, (lptr4_t)(v4i_t*)lds, 0, 0);
#else
  *(float4*)lds = *(const float4*)g;
#endif
}
__device__ __forceinline__ void wait_async() {
#if USE_ASYNC
  __builtin_amdgcn_s_wait_asynccnt(0);
#endif
}

#define LDX   268   // X/Q staging stride (256+12): 16B-aligned rows, bank-spread
#define LDB   264   // 32-row B tiles (256+8)
#define LDA_B 68    // K^T staging [32 tok][64 m] (64+4)

static constexpr int   CD = 256;
static constexpr int   HM = 1024;
static constexpr float NF = 32768.0f;

// stage a [32 x 256] row-major tile into sb (stride LDB); rs = global row stride
__device__ __forceinline__ void stage_w(float* sb, const float* g, size_t rs, int tid) {
  #pragma unroll
  for (int j = 0; j < 8; ++j) {
    int i = tid + j*256, row = i >> 6, seg = i & 63;
    cp16(sb + row*LDB + seg*4, g + (size_t)row*rs + seg*4);
  }
}

// ---------------------------------------------------------------------------
// Kernel A: Kn = l2norm_rows(X @ Wk_h), V = mask * (X @ Wv_h). 128-token tiles.
// grid (256, 4) x 256 thr. 8 waves = 2 rowgroups(64) x 4 colgroups(64);
// wave tile 64x64 => B fragments reused x4, A fragments reused x4.
// ---------------------------------------------------------------------------
extern "C" __global__ void __launch_bounds__(256)
k_proj_kv(const float* __restrict__ X, const float* __restrict__ Wk,
          const float* __restrict__ Wv, const unsigned char* __restrict__ msk,
          float* __restrict__ Kn, float* __restrict__ Vv)
{
  __shared__ float sX[128*LDX];
  __shared__ float sB[2][32*LDB];
  __shared__ float sRed[128*64];
  __shared__ float sScale[128];

  const int tid  = threadIdx.x;
  const int head = blockIdx.y;
  const int tok0 = blockIdx.x * 128;
  const int lane = tid & 31;
  const int wid  = tid >> 5;
  const int half = lane >> 4;
  const int kl   = half << 1;
  const int lc   = lane & 15;
  const int rg   = wid >> 2;
  const int cg   = wid & 3;
  const int rowbase = rg * 64;
  const int colbase = cg * 64;

  // Stage full X tile [128 x 256] once (async); reused for both GEMMs.
  #pragma unroll
  for (int j = 0; j < 32; ++j) {
    int i = tid + j*256, row = i >> 6, seg = i & 63;
    cp16(sX + row*LDX + seg*4, X + (size_t)(tok0+row)*CD + seg*4);
  }

  for (int mat = 0; mat < 2; ++mat) {
    const float* W = mat ? Wv : Wk;
    v8f acc[4][4];
    #pragma unroll
    for (int rt = 0; rt < 4; ++rt)
      #pragma unroll
      for (int ct = 0; ct < 4; ++ct) acc[rt][ct] = vz8();

    stage_w(sB[0], W + head*256, HM, tid);
    for (int kb = 0; kb < 8; ++kb) {
      wait_async();
      __syncthreads();
      if (kb < 7) stage_w(sB[(kb+1)&1], W + (size_t)(kb+1)*32*HM + head*256, HM, tid);
      const int buf = kb & 1;
      #pragma unroll
      for (int s = 0; s < 8; ++s) {
        const int kg = kb*32 + s*4 + kl;
        v2f a[4];
        #pragma unroll
        for (int rt = 0; rt < 4; ++rt)
          a[rt] = *(const v2f*)&sX[(rowbase + rt*16 + lc)*LDX + kg];
        #pragma unroll
        for (int ct = 0; ct < 4; ++ct) {
          const int cc = colbase + ct*16 + lc;
          v2f b;
          b.x = sB[buf][(s*4+kl  )*LDB + cc];
          b.y = sB[buf][(s*4+kl+1)*LDB + cc];
          #pragma unroll
          for (int rt = 0; rt < 4; ++rt) acc[rt][ct] = WMMA(a[rt], b, acc[rt][ct]);
        }
      }
    }

    if (mat == 0) {
      #pragma unroll
      for (int rt = 0; rt < 4; ++rt)
        #pragma unroll
        for (int i = 0; i < 8; ++i) {
          float s = 0.f;
          #pragma unroll
          for (int ct = 0; ct < 4; ++ct) { float v = acc[rt][ct][i]; s += v*v; }
          sRed[(rowbase + rt*16 + i + 8*half)*64 + cg*16 + lc] = s;
        }
      __syncthreads();
      if (tid < 128) {
        float s = 0.f;
        #pragma unroll
        for (int p = 0; p < 64; ++p) s += sRed[tid*64 + p];
        sScale[tid] = 1.0f / fmaxf(sqrtf(s), 1e-6f);
      }
      __syncthreads();
    } else {
      if (tid < 128) sScale[tid] = msk[tok0 + tid] ? 1.0f : 0.0f;
      __syncthreads();
    }

    float* dst = mat ? Vv : Kn;
    #pragma unroll
    for (int rt = 0; rt < 4; ++rt)
      #pragma unroll
      for (int ct = 0; ct < 4; ++ct)
        #pragma unroll
        for (int i = 0; i < 8; ++i) {
          int R  = rowbase + rt*16 + i + 8*half;
          int cc = colbase + ct*16 + lc;
          dst[(size_t)(tok0+R)*HM + head*256 + cc] = acc[rt][ct][i] * sScale[R];
        }
  }
}

// ---------------------------------------------------------------------------
// Kernel B: partial KV[h] += Kn_h^T @ V_h over 512-token chunks; partial ksum.
// grid (64 chunks, 4 m-groups, 4 heads) x 256 thr.
// 8 waves = 2 rowgroups(32 m) x 4 colgroups(64 d); wave tile 32x64.
// ---------------------------------------------------------------------------
extern "C" __global__ void __launch_bounds__(256)
k_kv(const float* __restrict__ Kn, const float* __restrict__ Vv,
     float* __restrict__ KVp, float* __restrict__ ksp)
{
  __shared__ float sA[2][32*LDA_B];
  __shared__ float sB[2][32*LDB];
  __shared__ float sKred[64*32];

  const int tid   = threadIdx.x;
  const int chunk = blockIdx.x;
  const int mg    = blockIdx.y;
  const int h     = blockIdx.z;
  const int t0    = chunk * 512;
  const int lane  = tid & 31;
  const int wid   = tid >> 5;
  const int half  = lane >> 4;
  const int kl    = half << 1;
  const int lc    = lane & 15;
  const int rg    = wid >> 2;
  const int cg    = wid & 3;

  float ksacc[2][4] = {{0.f,0.f,0.f,0.f},{0.f,0.f,0.f,0.f}};
  v8f acc[2][4];
  #pragma unroll
  for (int rt = 0; rt < 2; ++rt)
    #pragma unroll
    for (int ct = 0; ct < 4; ++ct) acc[rt][ct] = vz8();

  // register-path loader (values also feed ksum accumulation)
  auto loadA = [&](int buf, int it) {
    #pragma unroll
    for (int j = 0; j < 2; ++j) {
      int i = tid + j*256;
      int row = i >> 4, seg = i & 15;
      float4 v = *(const float4*)(Kn + (size_t)(t0 + it*32 + row)*HM + h*256 + mg*64 + seg*4);
      ksacc[j][0] += v.x; ksacc[j][1] += v.y; ksacc[j][2] += v.z; ksacc[j][3] += v.w;
      *(float4*)&sA[buf][row*LDA_B + seg*4] = v;
    }
  };

  loadA(0, 0);
  stage_w(sB[0], Vv + (size_t)t0*HM + h*256, HM, tid);
  for (int it = 0; it < 16; ++it) {
    wait_async();
    __syncthreads();
    if (it < 15) {
      loadA((it+1)&1, it+1);
      stage_w(sB[(it+1)&1], Vv + (size_t)(t0 + (it+1)*32)*HM + h*256, HM, tid);
    }
    const int buf = it & 1;
    #pragma unroll
    for (int s = 0; s < 8; ++s) {
      const int k0 = s*4 + kl;
      v2f a[2];
      #pragma unroll
      for (int rt = 0; rt < 2; ++rt) {
        a[rt].x = sA[buf][ k0   *LDA_B + rg*32 + rt*16 + lc];
        a[rt].y = sA[buf][(k0+1)*LDA_B + rg*32 + rt*16 + lc];
      }
      #pragma unroll
      for (int ct = 0; ct < 4; ++ct) {
        const int cc = cg*64 + ct*16 + lc;
        v2f b;
        b.x = sB[buf][ k0   *LDB + cc];
        b.y = sB[buf][(k0+1)*LDB + cc];
        #pragma unroll
        for (int rt = 0; rt < 2; ++rt) acc[rt][ct] = WMMA(a[rt], b, acc[rt][ct]);
      }
    }
  }

  const size_t base = ((size_t)((h*4 + mg)*64 + chunk)) * 16384;
  #pragma unroll
  for (int rt = 0; rt < 2; ++rt)
    #pragma unroll
    for (int ct = 0; ct < 4; ++ct)
      #pragma unroll
      for (int i = 0; i < 8; ++i) {
        int R  = rg*32 + rt*16 + i + 8*half;
        int cc = cg*64 + ct*16 + lc;
        KVp[base + (size_t)R*256 + cc] = acc[rt][ct][i];
      }

  #pragma unroll
  for (int j = 0; j < 2; ++j)
    #pragma unroll
    for (int c = 0; c < 4; ++c) {
      int m = (tid & 15)*4 + c;
      int p = (tid >> 4) + j*16;
      sKred[m*32 + p] = ksacc[j][c];
    }
  __syncthreads();
  if (tid < 64) {
    float s = 0.f;
    #pragma unroll
    for (int p = 0; p < 32; ++p) s += sKred[tid*32 + p];
    ksp[((size_t)((h*4 + mg)*64 + chunk))*64 + tid] = s;
  }
}

// ---------------------------------------------------------------------------
// Kernel R: reduce chunk partials -> KV[4][256][256], ksum[4][256]
// ---------------------------------------------------------------------------
extern "C" __global__ void __launch_bounds__(256)
k_reduce(const float* __restrict__ KVp, const float* __restrict__ ksp,
         float* __restrict__ KV, float* __restrict__ ksum)
{
  const int g = blockIdx.x*256 + threadIdx.x;
  #pragma unroll
  for (int r = 0; r < 4; ++r) {
    int o = g + r*65536;
    int h = o >> 16, rem = o & 65535, m = rem >> 8, d = rem & 255;
    int mg = m >> 6, ml = m & 63;
    const float* p = KVp + ((size_t)((h*4 + mg)*64))*16384 + (size_t)ml*256 + d;
    float s = 0.f;
    for (int c = 0; c < 64; ++c) s += p[(size_t)c*16384];
    KV[o] = s;
  }
  if (g < 1024) {
    int h = g >> 8, m = g & 255, mg = m >> 6, ml = m & 63;
    const float* p = ksp + ((size_t)((h*4 + mg)*64))*64 + ml;
    float s = 0.f;
    for (int c = 0; c < 64; ++c) s += p[c*64];
    ksum[g] = s;
  }
}

// ---------------------------------------------------------------------------
// Kernel C: Q proj + norm, den = q.ksum + n, num = Qn@KV + n*V, head-mean,
// residual avg, LayerNorm, ReLU. 64-token tiles. grid 512 x 256 thr.
// 8 waves = 2 rowgroups(32) x 4 colgroups(64); wave tile 32x64.
// ---------------------------------------------------------------------------
extern "C" __global__ void __launch_bounds__(256)
k_final(const float* __restrict__ X, const float* __restrict__ Wq,
        const float* __restrict__ KV, const float* __restrict__ ksum,
        const float* __restrict__ Vv, const float* __restrict__ LX,
        const float* __restrict__ lnw, const float* __restrict__ lnb,
        float* __restrict__ out)
{
  __shared__ float sX[64*LDX];
  __shared__ float sB[2][32*LDB];
  __shared__ float sQ[64*LDX];
  __shared__ float sRed[64*64];
  __shared__ float sMu[64];
  __shared__ float sScl[64];
  __shared__ float sKs[1024];
  __shared__ float sLw[256];
  __shared__ float sLb[256];

  const int tid  = threadIdx.x;
  const int tok0 = blockIdx.x * 64;
  const int lane = tid & 31;
  const int wid  = tid >> 5;
  const int half = lane >> 4;
  const int kl   = half << 1;
  const int lc   = lane & 15;
  const int rg   = wid >> 2;
  const int cg   = wid & 3;
  const int rowbase = rg * 32;

  #pragma unroll
  for (int j = 0; j < 16; ++j) {
    int i = tid + j*256, row = i >> 6, seg = i & 63;
    cp16(sX + row*LDX + seg*4, X + (size_t)(tok0+row)*CD + seg*4);
  }
  ((float4*)sKs)[tid] = ((const float4*)ksum)[tid];
  sLw[tid] = lnw[tid];
  sLb[tid] = lnb[tid];

  v8f att[2][4];
  #pragma unroll
  for (int rt = 0; rt < 2; ++rt)
    #pragma unroll
    for (int ct = 0; ct < 4; ++ct) att[rt][ct] = vz8();

  for (int h = 0; h < 4; ++h) {
    v8f acc[2][4];
    #pragma unroll
    for (int rt = 0; rt < 2; ++rt)
      #pragma unroll
      for (int ct = 0; ct < 4; ++ct) acc[rt][ct] = vz8();

    // ---- Q = X @ Wq_h ----
    stage_w(sB[0], Wq + h*256, HM, tid);
    for (int kb = 0; kb < 8; ++kb) {
      wait_async();
      __syncthreads();
      if (kb < 7) stage_w(sB[(kb+1)&1], Wq + (size_t)(kb+1)*32*HM + h*256, HM, tid);
      const int buf = kb & 1;
      #pragma unroll
      for (int s = 0; s < 8; ++s) {
        const int kg = kb*32 + s*4 + kl;
        v2f a[2];
        #pragma unroll
        for (int rt = 0; rt < 2; ++rt)
          a[rt] = *(const v2f*)&sX[(rowbase + rt*16 + lc)*LDX + kg];
        #pragma unroll
        for (int ct = 0; ct < 4; ++ct) {
          const int cc = cg*64 + ct*16 + lc;
          v2f b;
          b.x = sB[buf][(s*4+kl  )*LDB + cc];
          b.y = sB[buf][(s*4+kl+1)*LDB + cc];
          #pragma unroll
          for (int rt = 0; rt < 2; ++rt) acc[rt][ct] = WMMA(a[rt], b, acc[rt][ct]);
        }
      }
    }

    // ---- normalize Q rows into sQ ----
    #pragma unroll
    for (int rt = 0; rt < 2; ++rt)
      #pragma unroll
      for (int i = 0; i < 8; ++i) {
        float s = 0.f;
        #pragma unroll
        for (int ct = 0; ct < 4; ++ct) { float v = acc[rt][ct][i]; s += v*v; }
        sRed[(rowbase + rt*16 + i + 8*half)*64 + cg*16 + lc] = s;
      }
    __syncthreads();
    if (tid < 64) {
      float s = 0.f;
      #pragma unroll
      for (int p = 0; p < 64; ++p) s += sRed[tid*64 + p];
      sScl[tid] = 1.0f / fmaxf(sqrtf(s), 1e-6f);
    }
    __syncthreads();
    #pragma unroll
    for (int rt = 0; rt < 2; ++rt)
      #pragma unroll
      for (int ct = 0; ct < 4; ++ct)
        #pragma unroll
        for (int i = 0; i < 8; ++i) {
          int R  = rowbase + rt*16 + i + 8*half;
          int cc = cg*64 + ct*16 + lc;
          sQ[R*LDX + cc] = acc[rt][ct][i] * sScl[R];
        }
    __syncthreads();

    // ---- den[R] = qn . ksum_h + n ----
    {
      int row = tid >> 2, q = tid & 3;
      float s = 0.f;
      for (int jj = 0; jj < 64; ++jj)
        s += sQ[row*LDX + q*64 + jj] * sKs[h*256 + q*64 + jj];
      sRed[row*4 + q] = s;
    }
    __syncthreads();
    if (tid < 64)
      sMu[tid] = sRed[tid*4] + sRed[tid*4+1] + sRed[tid*4+2] + sRed[tid*4+3] + NF;
    __syncthreads();

    // ---- num = Qn @ KV_h ----
    #pragma unroll
    for (int rt = 0; rt < 2; ++rt)
      #pragma unroll
      for (int ct = 0; ct < 4; ++ct) acc[rt][ct] = vz8();
    stage_w(sB[0], KV + (size_t)h*65536, 256, tid);
    for (int kb = 0; kb < 8; ++kb) {
      wait_async();
      __syncthreads();
      if (kb < 7) stage_w(sB[(kb+1)&1], KV + (size_t)h*65536 + (size_t)(kb+1)*32*256, 256, tid);
      const int buf = kb & 1;
      #pragma unroll
      for (int s = 0; s < 8; ++s) {
        const int kg = kb*32 + s*4 + kl;
        v2f a[2];
        #pragma unroll
        for (int rt = 0; rt < 2; ++rt)
          a[rt] = *(const v2f*)&sQ[(rowbase + rt*16 + lc)*LDX + kg];
        #pragma unroll
        for (int ct = 0; ct < 4; ++ct) {
          const int cc = cg*64 + ct*16 + lc;
          v2f b;
          b.x = sB[buf][(s*4+kl  )*LDB + cc];
          b.y = sB[buf][(s*4+kl+1)*LDB + cc];
          #pragma unroll
          for (int rt = 0; rt < 2; ++rt) acc[rt][ct] = WMMA(a[rt], b, acc[rt][ct]);
        }
      }
    }

    // ---- att += (num + n*v) / den / H ----
    #pragma unroll
    for (int rt = 0; rt < 2; ++rt)
      #pragma unroll
      for (int ct = 0; ct < 4; ++ct)
        #pragma unroll
        for (int i = 0; i < 8; ++i) {
          int R  = rowbase + rt*16 + i + 8*half;
          int cc = cg*64 + ct*16 + lc;
          float v = Vv[(size_t)(tok0+R)*HM + h*256 + cc];
          att[rt][ct][i] += (acc[rt][ct][i] + NF*v) / sMu[R] * 0.25f;
        }
  }

  // ---- h = (att + last_x)/2 ----
  #pragma unroll
  for (int rt = 0; rt < 2; ++rt)
    #pragma unroll
    for (int ct = 0; ct < 4; ++ct)
      #pragma unroll
      for (int i = 0; i < 8; ++i) {
        int R = rowbase + rt*16 + i + 8*half;
        int c = cg*64 + ct*16 + lc;
        float lx = LX[(size_t)(tok0+R)*CD + c];
        att[rt][ct][i] = (att[rt][ct][i] + lx) * 0.5f;
      }

  // ---- LayerNorm: mean ----
  #pragma unroll
  for (int rt = 0; rt < 2; ++rt)
    #pragma unroll
    for (int i = 0; i < 8; ++i) {
      float s = 0.f;
      #pragma unroll
      for (int ct = 0; ct < 4; ++ct) s += att[rt][ct][i];
      sRed[(rowbase + rt*16 + i + 8*half)*64 + cg*16 + lc] = s;
    }
  __syncthreads();
  if (tid < 64) {
    float s = 0.f;
    #pragma unroll
    for (int p = 0; p < 64; ++p) s += sRed[tid*64 + p];
    sMu[tid] = s * (1.0f/256.0f);
  }
  __syncthreads();
  // ---- variance (centered) ----
  #pragma unroll
  for (int rt = 0; rt < 2; ++rt)
    #pragma unroll
    for (int i = 0; i < 8; ++i) {
      float m = sMu[rowbase + rt*16 + i + 8*half];
      float s = 0.f;
      #pragma unroll
      for (int ct = 0; ct < 4; ++ct) { float d = att[rt][ct][i] - m; s += d*d; }
      sRed[(rowbase + rt*16 + i + 8*half)*64 + cg*16 + lc] = s;
    }
  __syncthreads();
  if (tid < 64) {
    float s = 0.f;
    #pragma unroll
    for (int p = 0; p < 64; ++p) s += sRed[tid*64 + p];
    sScl[tid] = rsqrtf(s * (1.0f/256.0f) + 1e-5f);
  }
  __syncthreads();
  #pragma unroll
  for (int rt = 0; rt < 2; ++rt)
    #pragma unroll
    for (int ct = 0; ct < 4; ++ct)
      #pragma unroll
      for (int i = 0; i < 8; ++i) {
        int R = rowbase + rt*16 + i + 8*half;
        int c = cg*64 + ct*16 + lc;
        float o = (att[rt][ct][i] - sMu[R]) * sScl[R] * sLw[c] + sLb[c];
        out[(size_t)(tok0+R)*CD + c] = fmaxf(o, 0.0f);
      }
}

// ---------------------------------------------------------------------------
extern "C" void kernel_launch(void* const* d_in, const int* in_sizes, int n_in,
                              void* d_out, int out_size, void* d_ws, size_t ws_size,
                              hipStream_t stream)
{
  (void)in_sizes; (void)n_in; (void)out_size; (void)ws_size;
  const float* X   = (const float*)d_in[0];
  const float* LX  = (const float*)d_in[1];
  const float* Wq  = (const float*)d_in[2];
  const float* Wk  = (const float*)d_in[3];
  const float* Wv  = (const float*)d_in[4];
  const float* lnw = (const float*)d_in[5];
  const float* lnb = (const float*)d_in[6];
  const unsigned char* msk = (const unsigned char*)d_in[7];
  float* out = (float*)d_out;

  float* ws   = (float*)d_ws;
  float* Kn   = ws;                 // 32768*1024
  float* Vv   = ws + 33554432;      // 32768*1024
  float* KVp  = ws + 67108864;      // 4*4*64 * 64*256
  float* ksp  = ws + 83886080;      // 4*4*64 * 64
  float* KV   = ws + 83951616;      // 4*256*256
  float* ksum = ws + 84213760;      // 4*256

  hipLaunchKernelGGL(k_proj_kv, dim3(256, 4),    dim3(256), 0, stream, X, Wk, Wv, msk, Kn, Vv);
  hipLaunchKernelGGL(k_kv,      dim3(64, 4, 4),  dim3(256), 0, stream, Kn, Vv, KVp, ksp);
  hipLaunchKernelGGL(k_reduce,  dim3(256),       dim3(256), 0, stream, KVp, ksp, KV, ksum);
  hipLaunchKernelGGL(k_final,   dim3(512),       dim3(256), 0, stream, X, Wq, KV, ksum, Vv, LX, lnw, lnb, out);
}